// SUBNET_LPV_15290083574170
// MI455X (gfx1250) — compile-verified
//
#include <hip/hip_runtime.h>
#include <cstddef>

typedef float v2f __attribute__((ext_vector_type(2)));
typedef float v4f __attribute__((ext_vector_type(4)));
typedef float v8f __attribute__((ext_vector_type(8)));

#define T_STEPS 512

__device__ __forceinline__ v8f wmma_k4(v2f a, v2f b, v8f c) {
  // D = A(16x4 f32) * B(4x16 f32) + C(16x16 f32)
  return __builtin_amdgcn_wmma_f32_16x16x4_f32(false, a, false, b, (short)0, c, false, false);
}

__global__ __launch_bounds__(32) void lpv_subnet_scan(
    const float* __restrict__ upast, const float* __restrict__ ypast,
    const float* __restrict__ ufut,
    const float* __restrict__ eWres, const float* __restrict__ ebres,
    const float* __restrict__ eW1,  const float* __restrict__ eb1,
    const float* __restrict__ eW2,  const float* __restrict__ eb2,
    const float* __restrict__ eWo,  const float* __restrict__ ebo,
    const float* __restrict__ cWres, const float* __restrict__ cbres,
    const float* __restrict__ cW1,  const float* __restrict__ cb1,
    const float* __restrict__ cW2,  const float* __restrict__ cb2,
    const float* __restrict__ cWo,  const float* __restrict__ cbo,
    const float* __restrict__ A0, const float* __restrict__ Ap,
    const float* __restrict__ B0, const float* __restrict__ Bp,
    const float* __restrict__ C0, const float* __restrict__ Cp,
    const float* __restrict__ D0, const float* __restrict__ Dp,
    float* __restrict__ yout)
{
  // ---- persistent LDS (≈55 KB) ----
  __shared__ __attribute__((aligned(16))) float lW1[64*20];    // sch layer1  [m][k]
  __shared__ __attribute__((aligned(16))) float lW2[64*64];    // sch layer2
  __shared__ __attribute__((aligned(16))) float lWo[8*64];     // sch out
  __shared__ __attribute__((aligned(16))) float lWres[8*20];   // sch res
  __shared__ __attribute__((aligned(16))) float lG[20*180];    // [G0 | Gtilde], rows=20 outs, K=180
  __shared__ __attribute__((aligned(16))) float lB[16*180];    // staging: [n][ w(20) | p⊗w(160) ]
  __shared__ __attribute__((aligned(16))) float lH[16*64];     // h staging [n][64]
  __shared__ __attribute__((aligned(16))) float lP[16*8];      // p staging [n][8]
  __shared__ __attribute__((aligned(16))) float lb1[64];
  __shared__ __attribute__((aligned(16))) float lb2[64];
  __shared__ __attribute__((aligned(16))) float lpb[16];

  const int lane = threadIdx.x;
  const int n    = lane & 15;     // batch-in-tile / B-matrix column / A-matrix row
  const int hf   = lane >> 4;     // lane half selects K pair
  const int koff = 2 * hf;
  const int b0   = blockIdx.x * 16;

  // ================= fill persistent weights into LDS =================
  for (int i = lane; i < 64*20; i += 32) lW1[i]   = cW1[i];
  for (int i = lane; i < 64*64; i += 32) lW2[i]   = cW2[i];
  for (int i = lane; i < 8*64;  i += 32) lWo[i]   = cWo[i];
  for (int i = lane; i < 8*20;  i += 32) lWres[i] = cWres[i];
  for (int i = lane; i < 64;    i += 32) { lb1[i] = cb1[i]; lb2[i] = cb2[i]; }
  if (lane < 16) lpb[lane] = (lane < 8) ? (cbres[lane] + cbo[lane]) : 0.0f;
  for (int idx = lane; idx < 20*180; idx += 32) {
    int o = idx / 180, j = idx % 180; float v;
    if (j < 20) {                       // G0 = [[A0,B0],[C0,D0]]
      int i = j;
      v = (o < 16) ? ((i < 16) ? A0[o*16 + i] : B0[o*4 + (i-16)])
                   : ((i < 16) ? C0[(o-16)*16 + i] : D0[(o-16)*4 + (i-16)]);
    } else {                            // Gtilde[o][k*20+i] = Mk[o][i]
      int k = (j - 20) / 20, i = (j - 20) % 20;
      v = (o < 16) ? ((i < 16) ? Ap[k*256 + o*16 + i] : Bp[k*64 + o*4 + (i-16)])
                   : ((i < 16) ? Cp[k*64 + (o-16)*16 + i] : Dp[k*16 + (o-16)*4 + (i-16)]);
    }
    lG[idx] = v;
  }

  // ================= encoder (runs once) =================
  // stage z = [upast_flat | ypast_flat] into lB as [n][160] (aliased region)
  {
    const float* src = (hf == 0) ? (upast + (size_t)(b0 + n) * 80)
                                 : (ypast + (size_t)(b0 + n) * 80);
    int db = n*160 + hf*80;
    #pragma unroll
    for (int j = 0; j < 80; j += 4)
      *(v4f*)&lB[db + j] = *(const v4f*)&src[j];
  }
  // layer1: relu(eW1(64x160) @ z) -> lH [n][64]
  #pragma unroll 1
  for (int mt = 0; mt < 4; ++mt) {
    v8f c = {0,0,0,0,0,0,0,0};
    int row = mt*16 + n;
    for (int kc = 0; kc < 40; ++kc) {
      v2f a = *(const v2f*)&eW1[row*160 + kc*4 + koff];
      v2f b = *(const v2f*)&lB[n*160 + kc*4 + koff];
      c = wmma_k4(a, b, c);
    }
    v4f ba = *(const v4f*)&eb1[mt*16 + 8*hf];
    v4f bb = *(const v4f*)&eb1[mt*16 + 8*hf + 4];
    #pragma unroll
    for (int r = 0; r < 4; ++r) {
      c[r]   = fmaxf(c[r]   + ba[r], 0.0f);
      c[r+4] = fmaxf(c[r+4] + bb[r], 0.0f);
    }
    *(v8f*)&lH[n*64 + mt*16 + 8*hf] = c;   // contiguous m-run of 8
  }
  // layer2: relu(eW2 @ h1) -> back into lH (all reads precede stores)
  {
    v8f g[4];
    #pragma unroll 1
    for (int mt = 0; mt < 4; ++mt) {
      v8f c = {0,0,0,0,0,0,0,0};
      int row = mt*16 + n;
      for (int kc = 0; kc < 16; ++kc) {
        v2f a = *(const v2f*)&eW2[row*64 + kc*4 + koff];
        v2f b = *(const v2f*)&lH[n*64 + kc*4 + koff];
        c = wmma_k4(a, b, c);
      }
      v4f ba = *(const v4f*)&eb2[mt*16 + 8*hf];
      v4f bb = *(const v4f*)&eb2[mt*16 + 8*hf + 4];
      #pragma unroll
      for (int r = 0; r < 4; ++r) {
        c[r]   = fmaxf(c[r]   + ba[r], 0.0f);
        c[r+4] = fmaxf(c[r+4] + bb[r], 0.0f);
      }
      g[mt] = c;
    }
    #pragma unroll
    for (int mt = 0; mt < 4; ++mt)
      *(v8f*)&lH[n*64 + mt*16 + 8*hf] = g[mt];
  }
  // x0 = eWres @ z + eWo @ h2 + (ebres+ebo)  -> lB w-part [n][0..15]
  {
    v8f c = {0,0,0,0,0,0,0,0};
    for (int kc = 0; kc < 40; ++kc) {
      v2f a = *(const v2f*)&eWres[n*160 + kc*4 + koff];
      v2f b = *(const v2f*)&lB[n*160 + kc*4 + koff];
      c = wmma_k4(a, b, c);
    }
    for (int kc = 0; kc < 16; ++kc) {
      v2f a = *(const v2f*)&eWo[n*64 + kc*4 + koff];
      v2f b = *(const v2f*)&lH[n*64 + kc*4 + koff];
      c = wmma_k4(a, b, c);
    }
    v4f r1 = *(const v4f*)&ebres[8*hf];     v4f r2 = *(const v4f*)&ebres[8*hf + 4];
    v4f o1 = *(const v4f*)&ebo[8*hf];       v4f o2 = *(const v4f*)&ebo[8*hf + 4];
    v4f lo, hi;
    #pragma unroll
    for (int r = 0; r < 4; ++r) {
      lo[r] = c[r]   + r1[r] + o1[r];
      hi[r] = c[r+4] + r2[r] + o2[r];
    }
    *(v4f*)&lB[n*180 + 8*hf]     = lo;      // x rows m = 8*hf + 0..3
    *(v4f*)&lB[n*180 + 8*hf + 4] = hi;      // x rows m = 8*hf + 4..7
  }

  // ================= scan over T (sequential recurrence) =================
  const int prow  = (n < 8) ? n : 0;          // clamped A-row for 8-row p weights
  const int grow1 = 16 + ((n < 4) ? n : 0);   // clamped A-row for G tile 1 (rows 16..19 live)

  #pragma unroll 1
  for (int t = 0; t < T_STEPS; ++t) {
    // stream u_t into w[16..19]; prefetch next step's u
    if (lane < 16) {
      const float* up = &ufut[((size_t)(b0 + n) * T_STEPS + t) * 4];
      *(v4f*)&lB[n*180 + 16] = *(const v4f*)up;
      if (t + 1 < T_STEPS) __builtin_prefetch(up + 4, 0, 0);
    }
    // cache the 5 B-fragments of w (k = 0..19)
    v2f bw[5];
    #pragma unroll
    for (int kc = 0; kc < 5; ++kc)
      bw[kc] = *(const v2f*)&lB[n*180 + kc*4 + koff];

    // ---- sch layer1: relu(W1(64x20) @ w) -> lH ----
    #pragma unroll
    for (int mt = 0; mt < 4; ++mt) {
      v8f c = {0,0,0,0,0,0,0,0};
      int row = mt*16 + n;
      #pragma unroll
      for (int kc = 0; kc < 5; ++kc) {
        v2f a = *(const v2f*)&lW1[row*20 + kc*4 + koff];
        c = wmma_k4(a, bw[kc], c);
      }
      v4f ba = *(const v4f*)&lb1[mt*16 + 8*hf];
      v4f bb = *(const v4f*)&lb1[mt*16 + 8*hf + 4];
      #pragma unroll
      for (int r = 0; r < 4; ++r) {
        c[r]   = fmaxf(c[r]   + ba[r], 0.0f);
        c[r+4] = fmaxf(c[r+4] + bb[r], 0.0f);
      }
      *(v8f*)&lH[n*64 + mt*16 + 8*hf] = c;
    }
    // ---- sch layer2: relu(W2 @ h1) -> lH (reads before overwrites) ----
    {
      v8f g[4];
      #pragma unroll
      for (int mt = 0; mt < 4; ++mt) {
        v8f c = {0,0,0,0,0,0,0,0};
        int row = mt*16 + n;
        #pragma unroll
        for (int kc = 0; kc < 16; ++kc) {
          v2f a = *(const v2f*)&lW2[row*64 + kc*4 + koff];
          v2f b = *(const v2f*)&lH[n*64 + kc*4 + koff];
          c = wmma_k4(a, b, c);
        }
        v4f ba = *(const v4f*)&lb2[mt*16 + 8*hf];
        v4f bb = *(const v4f*)&lb2[mt*16 + 8*hf + 4];
        #pragma unroll
        for (int r = 0; r < 4; ++r) {
          c[r]   = fmaxf(c[r]   + ba[r], 0.0f);
          c[r+4] = fmaxf(c[r+4] + bb[r], 0.0f);
        }
        g[mt] = c;
      }
      #pragma unroll
      for (int mt = 0; mt < 4; ++mt)
        *(v8f*)&lH[n*64 + mt*16 + 8*hf] = g[mt];
    }
    // ---- p = Wres@w + Wo@h2 + (bres+bo); rows 8..15 are dont-care ----
    {
      v8f c = {0,0,0,0,0,0,0,0};
      #pragma unroll
      for (int kc = 0; kc < 5; ++kc) {
        v2f a = *(const v2f*)&lWres[prow*20 + kc*4 + koff];
        c = wmma_k4(a, bw[kc], c);
      }
      #pragma unroll
      for (int kc = 0; kc < 16; ++kc) {
        v2f a = *(const v2f*)&lWo[prow*64 + kc*4 + koff];
        v2f b = *(const v2f*)&lH[n*64 + kc*4 + koff];
        c = wmma_k4(a, b, c);
      }
      v4f pa = *(const v4f*)&lpb[8*hf];
      v4f pb = *(const v4f*)&lpb[8*hf + 4];
      #pragma unroll
      for (int r = 0; r < 4; ++r) { c[r] += pa[r]; c[r+4] += pb[r]; }
      if (lane < 16) *(v8f*)&lP[n*8] = c;   // lanes 0..15 hold p[0..7] for batch n
    }
    // ---- build p ⊗ w into lB[n][20..179] (VALU + DS) ----
    {
      v4f pv = *(const v4f*)&lP[n*8 + hf*4];      // this half handles k = 4*hf..4*hf+3
      v4f w0 = *(const v4f*)&lB[n*180 + 0];
      v4f w1 = *(const v4f*)&lB[n*180 + 4];
      v4f w2 = *(const v4f*)&lB[n*180 + 8];
      v4f w3 = *(const v4f*)&lB[n*180 + 12];
      v4f w4 = *(const v4f*)&lB[n*180 + 16];
      #pragma unroll
      for (int kk = 0; kk < 4; ++kk) {
        float pk = pv[kk];
        int base = n*180 + 20 + (hf*4 + kk)*20;
        *(v4f*)&lB[base + 0]  = pk * w0;
        *(v4f*)&lB[base + 4]  = pk * w1;
        *(v4f*)&lB[base + 8]  = pk * w2;
        *(v4f*)&lB[base + 12] = pk * w3;
        *(v4f*)&lB[base + 16] = pk * w4;
      }
    }
    // ---- dynamics: [xn; y] = [G0 | Gtilde] (20x180) @ [w; p⊗w] (180x16) ----
    v8f c0 = {0,0,0,0,0,0,0,0};   // rows 0..15  = xn
    v8f c1 = {0,0,0,0,0,0,0,0};   // rows 16..19 = y (rest dont-care)
    #pragma unroll 5
    for (int kc = 0; kc < 45; ++kc) {
      v2f b  = *(const v2f*)&lB[n*180 + kc*4 + koff];
      v2f a0 = *(const v2f*)&lG[n*180 + kc*4 + koff];
      v2f a1 = *(const v2f*)&lG[grow1*180 + kc*4 + koff];
      c0 = wmma_k4(a0, b, c0);
      c1 = wmma_k4(a1, b, c1);
    }
    // emit y_t
    if (lane < 16) {
      v4f y; y[0] = c1[0]; y[1] = c1[1]; y[2] = c1[2]; y[3] = c1[3];
      *(v4f*)&yout[((size_t)(b0 + n) * T_STEPS + t) * 4] = y;
    }
    // recycle xn into w-part for next step
    v4f xlo, xhi;
    #pragma unroll
    for (int r = 0; r < 4; ++r) { xlo[r] = c0[r]; xhi[r] = c0[r+4]; }
    *(v4f*)&lB[n*180 + 8*hf]     = xlo;
    *(v4f*)&lB[n*180 + 8*hf + 4] = xhi;
  }
}

extern "C" void kernel_launch(void* const* d_in, const int* in_sizes, int n_in,
                              void* d_out, int out_size, void* d_ws, size_t ws_size,
                              hipStream_t stream) {
  (void)in_sizes; (void)n_in; (void)out_size; (void)d_ws; (void)ws_size;
  lpv_subnet_scan<<<dim3(16), dim3(32), 0, stream>>>(
      (const float*)d_in[0],  (const float*)d_in[1],  (const float*)d_in[2],
      (const float*)d_in[4],  (const float*)d_in[5],  (const float*)d_in[6],
      (const float*)d_in[7],  (const float*)d_in[8],  (const float*)d_in[9],
      (const float*)d_in[10], (const float*)d_in[11],
      (const float*)d_in[12], (const float*)d_in[13], (const float*)d_in[14],
      (const float*)d_in[15], (const float*)d_in[16], (const float*)d_in[17],
      (const float*)d_in[18], (const float*)d_in[19],
      (const float*)d_in[20], (const float*)d_in[21], (const float*)d_in[22],
      (const float*)d_in[23], (const float*)d_in[24], (const float*)d_in[25],
      (const float*)d_in[26], (const float*)d_in[27],
      (float*)d_out);
}